// EnhancedInvariantExtractor_45973329936788
// MI455X (gfx1250) — compile-verified
//
#include <hip/hip_runtime.h>

typedef __attribute__((ext_vector_type(2))) float v2f;
typedef __attribute__((ext_vector_type(8))) float v8f;
typedef float vf3 __attribute__((ext_vector_type(3), aligned(4)));
typedef float vf4a __attribute__((ext_vector_type(4), aligned(4)));

#define IN_F  120
#define OUT_F 204
#define WAVES_PER_BLOCK 8

// Per-wave LDS region layout (float indices):
//   [0:16)    vnorm
//   [16:24)   tnorm
//   [24:280)  vgram 16x16 (clamped; lower triangle unused)
//   [280:536) tgram 16x16 (only top-left 8x8 meaningful)
#define R_VNORM 0
#define R_TNORM 16
#define R_VGRAM 24
#define R_TGRAM 280
#define R_SIZE  536   // 2144 B per wave, 16B multiple

// Gather table for output dwords 32..203 (172 entries = 43 int4 chunks).
struct SrcTable {
  alignas(16) int idx[172];
  constexpr SrcTable() : idx{} {
    int k = 0;
    for (int i = 0; i < 24; ++i) idx[k++] = i;  // vnorm(16) + tnorm(8)
    for (int i = 0; i < 16; ++i)
      for (int j = i + 1; j < 16; ++j) idx[k++] = R_VGRAM + i * 16 + j;
    for (int i = 0; i < 8; ++i)
      for (int j = i + 1; j < 8; ++j) idx[k++] = R_TGRAM + i * 16 + j;
  }
};
__constant__ const SrcTable c_tab = SrcTable();

// One wave32 per atom; grams via V_WMMA_F32_16X16X4_F32 (B regs == A regs for A·Aᵀ).
__global__ __launch_bounds__(256) void
invariant_extractor_kernel(const float* __restrict__ h,
                           float* __restrict__ out,
                           int n_atoms) {
  const int lane = threadIdx.x & 31;
  const int wave = threadIdx.x >> 5;
  const int atom = blockIdx.x * WAVES_PER_BLOCK + wave;

  __shared__ __align__(16) float stage[WAVES_PER_BLOCK][R_SIZE];
  float* sw = stage[wave];

  if (atom < n_atoms) {
    const float* __restrict__ ha = h + (size_t)atom * IN_F;
    float* __restrict__ op = out + (size_t)atom * OUT_F;

    const int top = lane >> 4;   // K-half: 0 -> K=0,1  1 -> K=2,3
    const int m   = lane & 15;   // matrix row handled by this lane

    // ---- scalars: direct coalesced 128B global->global copy ----
    op[lane] = ha[lane];

    // ---- vector rows (16 x 3): one b96 load, norm, normalized operand ----
    vf3 vr = *(const vf3*)(ha + 32 + 3 * m);
    float vx = vr.x, vy = vr.y, vz = vr.z;
    float vn = fmaxf(__builtin_amdgcn_sqrtf(vx * vx + vy * vy + vz * vz), 1e-6f);
    sw[R_VNORM + m] = vn;        // lanes m and m+16 write identical value
    float vinv = __builtin_amdgcn_rcpf(vn);
    v2f va;
    va.x = top ? vz * vinv : vx * vinv;   // K=2 / K=0
    va.y = top ? 0.0f      : vy * vinv;   // K=3(pad) / K=1

    // ---- tensor rows (8 x 5): b128 + b32 loads; rows 8-15 zeroed; K=5 -> 4+1 ----
    const float* trow = ha + 80 + 5 * (m & 7);
    vf4a tr = *(const vf4a*)trow;
    float t0 = tr.x, t1 = tr.y, t2 = tr.z, t3 = tr.w, t4 = trow[4];
    float tn = fmaxf(__builtin_amdgcn_sqrtf(t0*t0 + t1*t1 + t2*t2 + t3*t3 + t4*t4), 1e-6f);
    sw[R_TNORM + (m & 7)] = tn;  // 4 lanes write identical value
    float tinv = (m < 8) ? __builtin_amdgcn_rcpf(tn) : 0.0f;  // zero pad rows
    t0 *= tinv; t1 *= tinv; t2 *= tinv; t3 *= tinv; t4 *= tinv;
    v2f ta1, ta2;
    ta1.x = top ? t2   : t0;     // K=2 / K=0
    ta1.y = top ? t3   : t1;     // K=3 / K=1
    ta2.x = top ? 0.0f : t4;     // K=4 goes in second pass
    ta2.y = 0.0f;

    // ---- grams via WMMA ----
    v8f zero8 = {};
    v8f vg = __builtin_amdgcn_wmma_f32_16x16x4_f32(
        false, va, false, va, (short)0, zero8, false, false);
    v8f tg = __builtin_amdgcn_wmma_f32_16x16x4_f32(
        false, ta1, false, ta1, (short)0, zero8, false, false);
    tg = __builtin_amdgcn_wmma_f32_16x16x4_f32(
        false, ta2, false, ta2, (short)0, tg, false, false);

    // ---- branchless full-gram spill: one addr reg, imm offsets r*64 B ----
    float* vdst = sw + R_VGRAM + top * 128 + m;
    float* tdst = sw + R_TGRAM + top * 128 + m;
    #pragma unroll
    for (int r = 0; r < 8; ++r) {
      vdst[r * 16] = fminf(fmaxf(vg[r], -1.0f), 1.0f);
      tdst[r * 16] = fminf(fmaxf(tg[r], -1.0f), 1.0f);
    }
  }

  __syncthreads();

  if (atom < n_atoms) {
    // ---- gather 172 dwords via constant table, coalesced b128 stores ----
    float4* __restrict__ dst = (float4*)(out + (size_t)atom * OUT_F + 32);
    const int4* tab = (const int4*)c_tab.idx;
    #pragma unroll
    for (int c = lane; c < 43; c += 32) {
      int4 si = tab[c];
      float4 v;
      v.x = sw[si.x];
      v.y = sw[si.y];
      v.z = sw[si.z];
      v.w = sw[si.w];
      dst[c] = v;
    }
  }
}

extern "C" void kernel_launch(void* const* d_in, const int* in_sizes, int n_in,
                              void* d_out, int out_size, void* d_ws, size_t ws_size,
                              hipStream_t stream) {
  const float* h = (const float*)d_in[0];
  float* out = (float*)d_out;
  const int n_atoms = in_sizes[0] / IN_F;  // 1,000,000
  const int blocks = (n_atoms + WAVES_PER_BLOCK - 1) / WAVES_PER_BLOCK;
  invariant_extractor_kernel<<<blocks, 256, 0, stream>>>(h, out, n_atoms);
}